// TensorCPFactorization_386547057106
// MI455X (gfx1250) — compile-verified
//
#include <hip/hip_runtime.h>
#include <stdint.h>

// ---------------------------------------------------------------------------
// CP-factorization sampled reconstruction for MI455X (gfx1250).
//   out[b] = sum_r f0[i0[b],r] * f1[i1[b],r] * f2[i2[b],r],  RANK=32
// Strategy: stage all three 512x32 f32 factor matrices (192 KB total) into
// LDS with the Tensor Data Mover (tensor_load_to_lds, TENSORcnt), then do
// per-thread gathers from LDS (ds_load_b128) + VALU FMA reduction.
// This is gather/latency bound, not FLOP bound -> no WMMA; the CDNA5-specific
// paths used are TDM + s_wait_tensorcnt + 320KB-LDS staging.
// ---------------------------------------------------------------------------

typedef __attribute__((ext_vector_type(4))) unsigned int u32x4;
typedef __attribute__((ext_vector_type(8))) int          i32x8;
typedef __attribute__((ext_vector_type(4))) int          i32x4;
typedef __attribute__((ext_vector_type(4))) float        f32x4;

#define CP_RANK       32
#define CP_NROWS      512
#define MAT_ELEMS     (CP_NROWS * CP_RANK)   /* 16384 f32 */
#define MAT_BYTES     (MAT_ELEMS * 4)        /* 65536 B   */

// Issue one TDM load of a contiguous MAT_ELEMS-f32 block (global -> LDS).
// Descriptor layout per CDNA5 ISA sec. 8 (D# group 0 = 4 dwords, group 1 = 8
// dwords). 1-D tile: tensor_dim0 = tile_dim0 = 16384, tile_dim1 = 1.
__device__ __forceinline__ void tdm_load_mat_f32(const void* gsrc, uint32_t lds_byte_off) {
  const uint64_t ga = (uint64_t)(uintptr_t)gsrc;

  u32x4 g0;
  g0[0] = 1u;                                  // count=1 (valid), is_restore=0, gather off
  g0[1] = lds_byte_off;                        // lds_addr [63:32]
  g0[2] = (uint32_t)ga;                        // global_addr[31:0]  -> bits [95:64]
  g0[3] = ((uint32_t)(ga >> 32) & 0x01FFFFFFu) // global_addr[56:32] -> bits [120:96]
        | 0x80000000u;                         // type=2 ("image")   -> bits [127:126]

  i32x8 g1;
  g1[0] = 0x00020000;                          // workgroup_mask=0, data_size=2 (4 B), no flags
  g1[1] = (int)((MAT_ELEMS & 0xFFFFu) << 16);  // tensor_dim0[15:0] in bits [63:48]
  g1[2] = (int)(1u << 16);                     // tensor_dim0[31:16]=0, tensor_dim1=1
  g1[3] = (int)((MAT_ELEMS & 0xFFFFu) << 16);  // tile_dim0 = 16384 in bits [127:112]
  g1[4] = 1;                                   // tile_dim1=1, tile_dim2=0
  g1[5] = MAT_ELEMS;                           // tensor_dim0_stride low 32
  g1[6] = 0;                                   // stride0 hi / stride1 lo
  g1[7] = 0;

  const i32x4 zero4 = {0, 0, 0, 0};
  const i32x8 zero8 = {0, 0, 0, 0, 0, 0, 0, 0};
  // clang-23 / therock-10.0 6-arg form: (g0, g1, g2, g3, extra, cpol)
  __builtin_amdgcn_tensor_load_to_lds(g0, g1, zero4, zero4, zero8, 0);
}

__global__ __launch_bounds__(256) void
cp_sample_kernel(const float* __restrict__ f0,
                 const float* __restrict__ f1,
                 const float* __restrict__ f2,
                 const int*   __restrict__ idx32,   // int64 indices viewed as dword pairs
                 float*       __restrict__ out,
                 int batch) {
  extern __shared__ float smem[];                  // 3 * 64 KB = 192 KB (of 320 KB/WGP)
  float* s0 = smem;
  float* s1 = smem + MAT_ELEMS;
  float* s2 = smem + 2 * MAT_ELEMS;

  // Shared-aperture flat addresses keep the LDS byte offset in addr[31:0].
  const uint32_t lds_base = (uint32_t)(uintptr_t)smem;

  if (threadIdx.x < 32u) {                         // wave 0 drives the DMA
    tdm_load_mat_f32(f0, lds_base);
    tdm_load_mat_f32(f1, lds_base + MAT_BYTES);
    tdm_load_mat_f32(f2, lds_base + 2u * MAT_BYTES);
    __builtin_amdgcn_s_wait_tensorcnt(0);          // all three tiles resident
  }
  __syncthreads();                                 // publish LDS to all 8 waves

  const int stride = (int)(gridDim.x * blockDim.x);
  for (int b = (int)(blockIdx.x * blockDim.x + threadIdx.x); b < batch; b += stride) {
    // indices are int64 little-endian; values < 512 so the low dword suffices
    const int i0 = idx32[6 * b + 0];
    const int i1 = idx32[6 * b + 2];
    const int i2 = idx32[6 * b + 4];

    const f32x4* __restrict__ ra = (const f32x4*)(s0 + i0 * CP_RANK);
    const f32x4* __restrict__ rb = (const f32x4*)(s1 + i1 * CP_RANK);
    const f32x4* __restrict__ rc = (const f32x4*)(s2 + i2 * CP_RANK);

    float acc = 0.0f;
#pragma unroll
    for (int j = 0; j < CP_RANK / 4; ++j) {        // 3x ds_load_b128 per j
      const f32x4 a = ra[j], bq = rb[j], c = rc[j];
      acc = fmaf(a.x * bq.x, c.x, acc);
      acc = fmaf(a.y * bq.y, c.y, acc);
      acc = fmaf(a.z * bq.z, c.z, acc);
      acc = fmaf(a.w * bq.w, c.w, acc);
    }
    out[b] = acc;
  }
}

extern "C" void kernel_launch(void* const* d_in, const int* in_sizes, int n_in,
                              void* d_out, int out_size, void* d_ws, size_t ws_size,
                              hipStream_t stream) {
  const float* f0    = (const float*)d_in[0];
  const float* f1    = (const float*)d_in[1];
  const float* f2    = (const float*)d_in[2];
  const int*   idx32 = (const int*)d_in[3];        // int64 buffer viewed as int32 pairs
  float*       out   = (float*)d_out;
  const int    batch = out_size;                   // 65536

  const size_t shmem = 3u * (size_t)MAT_BYTES;     // 196608 bytes
  (void)hipFuncSetAttribute((const void*)cp_sample_kernel,
                            hipFuncAttributeMaxDynamicSharedMemorySize,
                            (int)shmem);

  // 128 blocks x 256 threads (8 wave32/block); 192 KB LDS -> 1 block per WGP.
  cp_sample_kernel<<<128, 256, shmem, stream>>>(f0, f1, f2, idx32, out, batch);
}